// SACLTCAgent_79164837200614
// MI455X (gfx1250) — compile-verified
//
#include <hip/hip_runtime.h>
#include <hip/hip_bf16.h>
#include <stdint.h>

#define B_TOT 4096
#define T_LEN 512
#define NF    8
#define HID   32
#define KH    16
#define NA    5
#define NBF   8

typedef __attribute__((ext_vector_type(2))) float v2f;
typedef __attribute__((ext_vector_type(8))) float v8f;

#define L2E 1.4426950408889634f
#define LN2 0.6931471805599453f

__device__ __forceinline__ float fexp2(float x){
#if __has_builtin(__builtin_amdgcn_exp2f)
  return __builtin_amdgcn_exp2f(x);
#else
  return exp2f(x);
#endif
}
__device__ __forceinline__ float flog2(float x){
#if __has_builtin(__builtin_amdgcn_logf)
  return __builtin_amdgcn_logf(x);
#else
  return log2f(x);
#endif
}
__device__ __forceinline__ float frcp(float x){
#if __has_builtin(__builtin_amdgcn_rcpf)
  return __builtin_amdgcn_rcpf(x);
#else
  return 1.0f / x;
#endif
}
__device__ __forceinline__ float ftanh(float x){
#if __has_builtin(__builtin_amdgcn_tanhf)
  return __builtin_amdgcn_tanhf(x);
#elif __has_builtin(__builtin_amdgcn_tanh_f32)
  return __builtin_amdgcn_tanh_f32(x);
#else
  float ax = __builtin_fabsf(x);
  float e  = fexp2(ax * (2.0f * L2E));
  float r  = 1.0f - 2.0f * frcp(e + 1.0f);
  return __builtin_copysignf(r, x);
#endif
}
__device__ __forceinline__ float fsoftplus(float x){
  float ax = __builtin_fabsf(x);
  return fmaxf(x, 0.0f) + flog2(1.0f + fexp2(-ax * L2E)) * LN2;
}

__device__ __forceinline__ v8f wmma4(v2f a, v2f b, v8f c){
  // D = A(16x4 f32) * B(4x16 f32) + C(16x16 f32)
  return __builtin_amdgcn_wmma_f32_16x16x4_f32(false, a, false, b, (short)0, c,
                                               false, false);
}
__device__ __forceinline__ v2f ld2(const float* p){
  v2f r; r.x = p[0]; r.y = p[1]; return r;
}

// LDS strides (floats), padded for conflict-free access across 16 rows
#define XS 12
#define HS 36
#define X_TILE (16 * XS)                 // one x staging buffer
#define WAVE_LDS (3 * X_TILE + 16 * HS)  // triple-buffered x + h stage

// ---------------------------------------------------------------------------
// Fused LTC scan: per wave, 16 batch rows, all 512 timesteps.
//   z    = bx   + Wx  @ x_t + Wh @ h      (transposed: D(M=hid,N=batch))
//   pre  = btau + Wtau@ x_t
//   tau  = tau_base + softplus(pre)       -> streamed to all_tau
//   h    = h + (tanh(z) - h) / tau
// x_t tiles are DMA'd global->LDS with the CDNA5 async path (ASYNCcnt),
// triple-buffered so the transfer overlaps the WMMA chain of the
// current step with no extra DS waits.
// ---------------------------------------------------------------------------
__global__ void __launch_bounds__(64) ltc_scan_kernel(
    const float* __restrict__ x_seq, const float* __restrict__ Wx,
    const float* __restrict__ bx,    const float* __restrict__ Wh,
    const float* __restrict__ Wtau,  const float* __restrict__ btau,
    const float* __restrict__ tau_base,
    float* __restrict__ h_final, float* __restrict__ all_tau)
{
  __shared__ __align__(16) float lds[2 * WAVE_LDS];
  const int wid  = threadIdx.x >> 5;
  const int lane = threadIdx.x & 31;
  const int hi   = lane >> 4;     // lane half (selects K/M sub-blocks)
  const int lp   = lane & 15;     // position within half (N = batch row)
  const int batch0 = (blockIdx.x * 2 + wid) * 16;

  float* ldsx = lds + wid * WAVE_LDS;       // 3 x-buffers of X_TILE floats
  float* ldsh = ldsx + 3 * X_TILE;          // h stage

  // Stationary A operands (weights) in WMMA A-layout:
  // vgpr j of tile(mt,kb): lane -> W[mt*16 + lp][kb*4 + 2*hi + j]
  v2f a_wx[2][2], a_wt[2][2], a_wh[2][8];
  #pragma unroll
  for (int mt = 0; mt < 2; ++mt) {
    const int m = mt * 16 + lp;
    #pragma unroll
    for (int kb = 0; kb < 2; ++kb) {
      const int k = kb * 4 + 2 * hi;
      a_wx[mt][kb] = ld2(Wx   + m * NF + k);
      a_wt[mt][kb] = ld2(Wtau + m * NF + k);
    }
    #pragma unroll
    for (int kb = 0; kb < 8; ++kb) {
      const int k = kb * 4 + 2 * hi;
      a_wh[mt][kb] = ld2(Wh + m * HID + k);
    }
  }

  // Biases / tau_base pre-swizzled into C/D layout: elem r -> m = mt*16+hi*8+r
  v8f vbx[2], vbt[2], vtb[2];
  #pragma unroll
  for (int mt = 0; mt < 2; ++mt)
    #pragma unroll
    for (int r = 0; r < 8; ++r) {
      const int m = mt * 16 + hi * 8 + r;
      vbx[mt][r] = bx[m];
      vbt[mt][r] = btau[m];
      vtb[mt][r] = tau_base[m];
    }

  v8f h0 = {}, h1 = {};   // h kept in WMMA C/D layout (M=hid, N=batch)

  // Async x staging: each lane owns one 16B chunk of the (16 x 8) x_t tile.
  const int xrow = lane >> 1, xhalf = lane & 1;
  const float* xptr = x_seq + (size_t)(batch0 + xrow) * T_LEN * NF + xhalf * 4;
  const int ldsx_lane_off = xrow * XS + xhalf * 4;

  // Kick off the DMA for t = 0 into buffer 0.
  {
    uint32_t dst = (uint32_t)(uintptr_t)&ldsx[ldsx_lane_off];
    uint64_t ga  = (uint64_t)(uintptr_t)xptr;
    asm volatile("global_load_async_to_lds_b128 %0, %1, off"
                 :: "v"(dst), "v"(ga) : "memory");
  }

  int xcur = 0, xnxt = 1;
  for (int t = 0; t < T_LEN; ++t) {
    // ---- stage h into wave-private LDS (canonical row-major) ----
    #pragma unroll
    for (int mt = 0; mt < 2; ++mt) {
      const v8f hv = mt ? h1 : h0;
      float* hrow = &ldsh[lp * HS + mt * 16 + hi * 8];
      *(float4*)(hrow + 0) = make_float4(hv[0], hv[1], hv[2], hv[3]);
      *(float4*)(hrow + 4) = make_float4(hv[4], hv[5], hv[6], hv[7]);
    }

    // ---- async DMA of x_{t+1} into the next buffer; wait for x_t ----
    if (t + 1 < T_LEN) {
      uint32_t dst = (uint32_t)(uintptr_t)&ldsx[xnxt * X_TILE + ldsx_lane_off];
      uint64_t ga  = (uint64_t)(uintptr_t)(xptr + (size_t)(t + 1) * NF);
      asm volatile("global_load_async_to_lds_b128 %0, %1, off"
                   :: "v"(dst), "v"(ga) : "memory");
      asm volatile("s_wait_asynccnt 0x1" ::: "memory");
    } else {
      asm volatile("s_wait_asynccnt 0x0" ::: "memory");
    }
    if (t + 2 < T_LEN)
      __builtin_prefetch(xptr + (size_t)(t + 2) * NF, 0, 1);
    __builtin_amdgcn_wave_barrier();

    // ---- WMMA: z = bx + Wx@x + Wh@h ; pre = btau + Wtau@x ----
    // z split into two independent half-chains for ILP (max depth 6 vs 10).
    const float* xb = ldsx + xcur * X_TILE;
    v8f z0a = vbx[0], z1a = vbx[1], p0 = vbt[0], p1 = vbt[1];
    v8f z0b = {}, z1b = {};
    #pragma unroll
    for (int kb = 0; kb < 2; ++kb) {              // K over 8 features
      const v2f b = ld2(&xb[lp * XS + kb * 4 + 2 * hi]);
      z0a = wmma4(a_wx[0][kb], b, z0a);
      z1a = wmma4(a_wx[1][kb], b, z1a);
      p0  = wmma4(a_wt[0][kb], b, p0);
      p1  = wmma4(a_wt[1][kb], b, p1);
    }
    #pragma unroll
    for (int kb = 0; kb < 4; ++kb) {              // K over hidden 0..15
      const v2f b = ld2(&ldsh[lp * HS + kb * 4 + 2 * hi]);
      z0a = wmma4(a_wh[0][kb], b, z0a);
      z1a = wmma4(a_wh[1][kb], b, z1a);
    }
    #pragma unroll
    for (int kb = 4; kb < 8; ++kb) {              // K over hidden 16..31
      const v2f b = ld2(&ldsh[lp * HS + kb * 4 + 2 * hi]);
      z0b = wmma4(a_wh[0][kb], b, z0b);
      z1b = wmma4(a_wh[1][kb], b, z1b);
    }
    __builtin_amdgcn_wave_barrier();
    const v8f z0 = z0a + z0b;
    const v8f z1 = z1a + z1b;

    // ---- elementwise LTC update, directly in C/D layout ----
    v8f tau0, tau1;
    #pragma unroll
    for (int r = 0; r < 8; ++r) {
      const float f0 = ftanh(z0[r]);
      const float f1 = ftanh(z1[r]);
      const float t0 = vtb[0][r] + fsoftplus(p0[r]);
      const float t1 = vtb[1][r] + fsoftplus(p1[r]);
      h0[r] += (f0 - h0[r]) * frcp(t0);
      h1[r] += (f1 - h1[r]) * frcp(t1);
      tau0[r] = t0; tau1[r] = t1;
    }

    // ---- stream tau to all_tau[t][batch][hid] (b128 stores) ----
    float* tp = all_tau + ((size_t)t * B_TOT + batch0 + lp) * HID + hi * 8;
    *(float4*)(tp +  0) = make_float4(tau0[0], tau0[1], tau0[2], tau0[3]);
    *(float4*)(tp +  4) = make_float4(tau0[4], tau0[5], tau0[6], tau0[7]);
    *(float4*)(tp + 16) = make_float4(tau1[0], tau1[1], tau1[2], tau1[3]);
    *(float4*)(tp + 20) = make_float4(tau1[4], tau1[5], tau1[6], tau1[7]);

    // rotate x triple-buffer
    xcur = xnxt;
    xnxt = xnxt + 1;
    if (xnxt == 3) xnxt = 0;
  }

  float* hp = h_final + (size_t)(batch0 + lp) * HID + hi * 8;
  *(float4*)(hp +  0) = make_float4(h0[0], h0[1], h0[2], h0[3]);
  *(float4*)(hp +  4) = make_float4(h0[4], h0[5], h0[6], h0[7]);
  *(float4*)(hp + 16) = make_float4(h1[0], h1[1], h1[2], h1[3]);
  *(float4*)(hp + 20) = make_float4(h1[4], h1[5], h1[6], h1[7]);
}

// ---------------------------------------------------------------------------
// KAN1 -> LayerNorm -> KAN2 -> softmax head. One thread per batch row.
// ---------------------------------------------------------------------------
__global__ void __launch_bounds__(256) kan_head_kernel(
    const float* __restrict__ hfin,
    const float* __restrict__ k1_bw, const float* __restrict__ k1_bb,
    const float* __restrict__ k1_sw,
    const float* __restrict__ ln_g,  const float* __restrict__ ln_b,
    const float* __restrict__ k2_bw, const float* __restrict__ k2_bb,
    const float* __restrict__ k2_sw,
    const float* __restrict__ grid,  float* __restrict__ probs)
{
  const int row = blockIdx.x * blockDim.x + threadIdx.x;
  if (row >= B_TOT) return;
  const float* h = hfin + (size_t)row * HID;

  float g[NBF];
  #pragma unroll
  for (int j = 0; j < NBF; ++j) g[j] = grid[j];

  float acc[KH];
  #pragma unroll
  for (int o = 0; o < KH; ++o) acc[o] = k1_bb[o];

  for (int i = 0; i < HID; ++i) {
    const float x = h[i];
    const float s = x * frcp(1.0f + fexp2(-x * L2E));   // SiLU
    float bas[NBF];
    #pragma unroll
    for (int j = 0; j < NBF; ++j) {
      const float d = (x - g[j]) * (1.0f / 0.3f);
      bas[j] = fexp2(-0.5f * L2E * d * d);              // Gaussian RBF
    }
    #pragma unroll
    for (int o = 0; o < KH; ++o) {
      float a = acc[o] + s * k1_bw[o * HID + i];
      #pragma unroll
      for (int j = 0; j < NBF; ++j)
        a += bas[j] * k1_sw[(i * NBF + j) * KH + o];
      acc[o] = a;
    }
  }

  float mu = 0.f;
  #pragma unroll
  for (int o = 0; o < KH; ++o) mu += acc[o];
  mu *= (1.0f / KH);
  float var = 0.f;
  #pragma unroll
  for (int o = 0; o < KH; ++o) { const float d = acc[o] - mu; var += d * d; }
  var *= (1.0f / KH);
  const float rs = rsqrtf(var + 1e-5f);
  float y[KH];
  #pragma unroll
  for (int o = 0; o < KH; ++o) y[o] = (acc[o] - mu) * rs * ln_g[o] + ln_b[o];

  float l[NA];
  #pragma unroll
  for (int a = 0; a < NA; ++a) l[a] = k2_bb[a];
  #pragma unroll
  for (int i = 0; i < KH; ++i) {
    const float x = y[i];
    const float s = x * frcp(1.0f + fexp2(-x * L2E));
    float bas[NBF];
    #pragma unroll
    for (int j = 0; j < NBF; ++j) {
      const float d = (x - g[j]) * (1.0f / 0.3f);
      bas[j] = fexp2(-0.5f * L2E * d * d);
    }
    #pragma unroll
    for (int a = 0; a < NA; ++a) {
      float v = l[a] + s * k2_bw[a * KH + i];
      #pragma unroll
      for (int j = 0; j < NBF; ++j)
        v += bas[j] * k2_sw[(i * NBF + j) * NA + a];
      l[a] = v;
    }
  }

  float m = l[0];
  #pragma unroll
  for (int a = 1; a < NA; ++a) m = fmaxf(m, l[a]);
  float e[NA], sum = 0.f;
  #pragma unroll
  for (int a = 0; a < NA; ++a) { e[a] = fexp2((l[a] - m) * L2E); sum += e[a]; }
  const float inv = frcp(sum);
  #pragma unroll
  for (int a = 0; a < NA; ++a) probs[(size_t)row * NA + a] = e[a] * inv;
}

// ---------------------------------------------------------------------------
extern "C" void kernel_launch(void* const* d_in, const int* in_sizes, int n_in,
                              void* d_out, int out_size, void* d_ws, size_t ws_size,
                              hipStream_t stream) {
  (void)in_sizes; (void)n_in; (void)out_size; (void)d_ws; (void)ws_size;

  const float* x_seq = (const float*)d_in[0];
  const float* Wx    = (const float*)d_in[1];
  const float* bx    = (const float*)d_in[2];
  const float* Wh    = (const float*)d_in[3];
  const float* Wtau  = (const float*)d_in[4];
  const float* btau  = (const float*)d_in[5];
  const float* tb    = (const float*)d_in[6];
  const float* k1_bw = (const float*)d_in[7];
  const float* k1_bb = (const float*)d_in[8];
  const float* k1_sw = (const float*)d_in[9];
  const float* ln_g  = (const float*)d_in[10];
  const float* ln_b  = (const float*)d_in[11];
  const float* k2_bw = (const float*)d_in[12];
  const float* k2_bb = (const float*)d_in[13];
  const float* k2_sw = (const float*)d_in[14];
  const float* grid  = (const float*)d_in[15];

  float* probs   = (float*)d_out;                       // (4096, 5)
  float* h_final = probs + (size_t)B_TOT * NA;          // (4096, 32)
  float* all_tau = h_final + (size_t)B_TOT * HID;       // (512, 4096, 32)

  ltc_scan_kernel<<<dim3(B_TOT / 32), dim3(64), 0, stream>>>(
      x_seq, Wx, bx, Wh, Wtau, btau, tb, h_final, all_tau);
  kan_head_kernel<<<dim3(B_TOT / 256), dim3(256), 0, stream>>>(
      h_final, k1_bw, k1_bb, k1_sw, ln_g, ln_b, k2_bw, k2_bb, k2_sw, grid,
      probs);
}